// FilterByOctaves_11072425689358
// MI455X (gfx1250) — compile-verified
//
#include <hip/hip_runtime.h>
#include <hip/hip_bf16.h>

// ---------------------------------------------------------------------------
// Octave filter bank (5th-order Butterworth bandpass cascade, DF2T) on MI455X.
//
// Exact block state-space reformulation, chunk L=16:
//   [Y ; Z'] = [Gy ; Gz] (16x32 each, per band) x W(32x16),
//   W = [ X_chunk (16 samples) ; z-state (10, slot-permuted) ; 0 (6) ],
// over 16 batches per wave, with chained V_WMMA_F32_16X16X4_F32 (f32 end to
// end: pole radii ~0.994..0.999 make half-precision unusable).
//
// Latency fix (the chain, not bandwidth, is the bottleneck): segmented
// two-pass scan over P=60 segments of 125 chunks:
//   pass1  (7 bands x 2 groups x 60 segs = 840 waves): state-only recurrence
//          from z=0 -> per-segment input response F_p (stored in D layout).
//   combine (14 waves, 60 steps): z_start[p+1] = A^Tseg * z_start[p] + F_p
//          via 3 chained WMMAs per step (A^Tseg slot-permuted, C = F_p).
//   pass2  (840 waves): full recurrence with exact initial state, writes Y.
// x (15 MB) lives in the 192 MB L2 across both passes and all 7 bands, so
// HBM traffic stays ~15 MB in + 107 MB out (~5.3 us floor at 23.3 TB/s).
// ---------------------------------------------------------------------------

typedef float v2f __attribute__((ext_vector_type(2)));
typedef float v4f __attribute__((ext_vector_type(4)));
typedef float v8f __attribute__((ext_vector_type(8)));

#define NBANDS 7
#define NSEC   5
#define NT     120000
#define CH     16
#define NCHUNK (NT / CH)            // 7500
#define NSEG   60
#define CSEG   (NCHUNK / NSEG)      // 125 chunks / segment
#define TSEG   (CSEG * CH)          // 2000 samples / segment
#define GSTRIDE 512                 // 16*32 floats per block operator

// workspace layout (in floats)
#define OFF_GY 0
#define OFF_GZ (OFF_GY + NBANDS * GSTRIDE)
#define OFF_GS (OFF_GZ + NBANDS * GSTRIDE)          // A^Tseg, slot-permuted
#define OFF_F  (OFF_GS + NBANDS * GSTRIDE)          // [14][NSEG][256]
#define OFF_ZS (OFF_F + NBANDS * 2 * NSEG * 256)    // [14][NSEG][256]

// state-slot permutation: state i lives at D row slot[i] so its half (row/8)
// matches the B-operand lane half that consumes it next chunk.
__device__ __constant__ int kSlot[10] = {0, 1, 8, 9, 2, 3, 10, 11, 4, 5};

// ---------------------------------------------------------------------------
// Setup: per band, build Gy / Gz (16-sample block operators) and Gs = A^Tseg
// by exact simulation of the cascade on basis vectors. One column per thread.
// ---------------------------------------------------------------------------
__global__ __launch_bounds__(32)
void build_block_ops(const float* __restrict__ sos, float* __restrict__ G) {
  const int band = blockIdx.x;
  const int c    = threadIdx.x;
  float* Gy = G + OFF_GY + band * GSTRIDE;
  float* Gz = G + OFF_GZ + band * GSTRIDE;
  float* Gs = G + OFF_GS + band * GSTRIDE;
  for (int r = 0; r < 16; ++r) {
    Gy[r * 32 + c] = 0.f; Gz[r * 32 + c] = 0.f; Gs[r * 32 + c] = 0.f;
  }
  if (c < CH + 2 * NSEC) {  // 26 basis columns: 16 impulses + 10 state units
    float b0[NSEC], b1[NSEC], b2[NSEC], a1[NSEC], a2[NSEC];
    for (int i = 0; i < NSEC; ++i) {
      const float* r = sos + (band * NSEC + i) * 6;
      b0[i] = r[0]; b1[i] = r[1]; b2[i] = r[2]; a1[i] = r[4]; a2[i] = r[5];
    }
    float z[2 * NSEC];
    for (int i = 0; i < 2 * NSEC; ++i) z[i] = 0.f;
    if (c >= CH) z[c - CH] = 1.f;

    for (int m = 0; m < CH; ++m) {            // 16 DF2T cascade steps
      float s = (c < CH && m == c) ? 1.f : 0.f;
      for (int i = 0; i < NSEC; ++i) {
        float y      = b0[i] * s + z[2 * i];
        z[2 * i]     = b1[i] * s - a1[i] * y + z[2 * i + 1];
        z[2 * i + 1] = b2[i] * s - a2[i] * y;
        s = y;
      }
      Gy[m * 32 + c] = s;
    }
    for (int i = 0; i < 2 * NSEC; ++i) Gz[kSlot[i] * 32 + c] = z[i];

    // homogeneous propagation continues to Tseg steps -> column of A^Tseg
    if (c >= CH) {
      for (int m = CH; m < TSEG; ++m) {
        float s = 0.f;
        for (int i = 0; i < NSEC; ++i) {
          float y      = b0[i] * s + z[2 * i];
          z[2 * i]     = b1[i] * s - a1[i] * y + z[2 * i + 1];
          z[2 * i + 1] = b2[i] * s - a2[i] * y;
          s = y;
        }
      }
      for (int i = 0; i < 2 * NSEC; ++i) Gs[kSlot[i] * 32 + c] = z[i];
    }
  }
}

// ---------------------------------------------------------------------------
static __device__ __forceinline__ v8f wmma_f32(v2f a, v2f b, v8f c) {
  return __builtin_amdgcn_wmma_f32_16x16x4_f32(
      false, a, false, b, (short)0, c, false, false);
}
static __device__ __forceinline__ void store_zd(float* p, v8f z) {
  v4f lo = { z[0], z[1], z[2], z[3] };
  v4f hi = { z[4], z[5], z[6], z[7] };
  *(v4f*)(p)     = lo;
  *(v4f*)(p + 4) = hi;
}
static __device__ __forceinline__ v8f load_zd(const float* p) {
  v4f lo = *(const v4f*)(p);
  v4f hi = *(const v4f*)(p + 4);
  v8f z = { lo[0], lo[1], lo[2], lo[3], hi[0], hi[1], hi[2], hi[3] };
  return z;
}

// ---------------------------------------------------------------------------
// Segment pass. WRITE_Y=false: pass 1 (state only, from z=0, emit F_p).
//               WRITE_Y=true : pass 2 (full, from z_start[p], emit Y).
// grid = NBANDS*2*NSEG single-wave blocks.
// ---------------------------------------------------------------------------
template <bool WRITE_Y>
__global__ __launch_bounds__(32)
void iir_seg_pass(const float* __restrict__ x, float* __restrict__ G,
                  float* __restrict__ out) {
  const int seg  = blockIdx.x % NSEG;
  const int pair = blockIdx.x / NSEG;         // band*2 + grp
  const int band = pair >> 1;
  const int grp  = pair & 1;
  const int lane = threadIdx.x;
  const int h    = lane >> 4;
  const int m    = lane & 15;

  // A-operand slices (A layout: lane 16h+m, vgpr v -> G[m][4j+2h+v])
  const float* Gy = G + OFF_GY + band * GSTRIDE;
  const float* Gz = G + OFF_GZ + band * GSTRIDE;
  v2f Ay[7], Az[7];
#pragma unroll
  for (int j = 0; j < 7; ++j) {
    const int col = 4 * j + 2 * h;
    if (WRITE_Y) { Ay[j][0] = Gy[m * 32 + col]; Ay[j][1] = Gy[m * 32 + col + 1]; }
    Az[j][0] = Gz[m * 32 + col]; Az[j][1] = Gz[m * 32 + col + 1];
  }

  const float* xrow = x + (size_t)(16 * grp + m) * NT;
  float* orow = out + ((size_t)(16 * grp + m) * NBANDS + band) * NT;

  v8f Zd;
  if (WRITE_Y) Zd = load_zd(G + OFF_ZS + (pair * NSEG + seg) * 256 + lane * 8);
  else         Zd = v8f{};

  const int c0 = seg * CSEG;
  for (int c = c0; c < c0 + CSEG; ++c) {
    const int tb = c * CH;
    if (tb + 16 * CH < NT)
      __builtin_prefetch(xrow + tb + 16 * CH, 0, 3);   // WGP-scope stream-ahead

    // X B-slices (B layout: lane 16h+n, vgpr v -> W[4j+2h+v][n])
    v2f Bx0 = *(const v2f*)(xrow + tb + 0  + 2 * h);
    v2f Bx1 = *(const v2f*)(xrow + tb + 4  + 2 * h);
    v2f Bx2 = *(const v2f*)(xrow + tb + 8  + 2 * h);
    v2f Bx3 = *(const v2f*)(xrow + tb + 12 + 2 * h);
    // state B-slices: register renames of previous D (slot permutation)
    v2f Bz4 = { Zd[0], Zd[1] };
    v2f Bz5 = { Zd[2], Zd[3] };
    v2f Bz6 = { h ? 0.f : Zd[4], h ? 0.f : Zd[5] };

    v8f zn = {};
    zn = wmma_f32(Az[0], Bx0, zn);
    zn = wmma_f32(Az[1], Bx1, zn);
    zn = wmma_f32(Az[2], Bx2, zn);
    zn = wmma_f32(Az[3], Bx3, zn);
    if (WRITE_Y) {
      v8f y = {};
      y = wmma_f32(Ay[0], Bx0, y);
      y = wmma_f32(Ay[1], Bx1, y);
      y = wmma_f32(Ay[2], Bx2, y);
      y = wmma_f32(Ay[3], Bx3, y);
      y = wmma_f32(Ay[4], Bz4, y);
      y = wmma_f32(Ay[5], Bz5, y);
      y = wmma_f32(Ay[6], Bz6, y);
      store_zd(orow + tb + 8 * h, y);       // two global_store_b128
    }
    zn = wmma_f32(Az[4], Bz4, zn);          // the true 3-WMMA recurrence
    zn = wmma_f32(Az[5], Bz5, zn);
    zn = wmma_f32(Az[6], Bz6, zn);
    Zd = zn;
  }
  if (!WRITE_Y)                              // per-segment input response F_p
    store_zd(G + OFF_F + (pair * NSEG + seg) * 256 + lane * 8, Zd);
}

// ---------------------------------------------------------------------------
// Combine: z_start[p+1] = A^Tseg * z_start[p] + F_p  (3 chained WMMAs / step)
// grid = 14 single-wave blocks.
// ---------------------------------------------------------------------------
__global__ __launch_bounds__(32)
void combine_segments(float* __restrict__ G) {
  const int pair = blockIdx.x;
  const int band = pair >> 1;
  const int lane = threadIdx.x;
  const int h    = lane >> 4;
  const int m    = lane & 15;

  const float* Gs = G + OFF_GS + band * GSTRIDE;
  v2f As[3];
#pragma unroll
  for (int j = 0; j < 3; ++j) {     // slices j=4,5,6 (state columns 16..27)
    const int col = 4 * (j + 4) + 2 * h;
    As[j][0] = Gs[m * 32 + col]; As[j][1] = Gs[m * 32 + col + 1];
  }

  v8f Zs = {};
  for (int p = 0; p < NSEG; ++p) {
    store_zd(G + OFF_ZS + (pair * NSEG + p) * 256 + lane * 8, Zs);
    v8f acc = load_zd(G + OFF_F + (pair * NSEG + p) * 256 + lane * 8);
    v2f Bz4 = { Zs[0], Zs[1] };
    v2f Bz5 = { Zs[2], Zs[3] };
    v2f Bz6 = { h ? 0.f : Zs[4], h ? 0.f : Zs[5] };
    acc = wmma_f32(As[0], Bz4, acc);
    acc = wmma_f32(As[1], Bz5, acc);
    acc = wmma_f32(As[2], Bz6, acc);
    Zs = acc;
  }
}

// ---------------------------------------------------------------------------
extern "C" void kernel_launch(void* const* d_in, const int* in_sizes, int n_in,
                              void* d_out, int out_size, void* d_ws, size_t ws_size,
                              hipStream_t stream) {
  (void)in_sizes; (void)n_in; (void)out_size; (void)ws_size;
  const float* x   = (const float*)d_in[0];  // [32][120000] f32
  const float* sos = (const float*)d_in[1];  // [7][5][6]    f32
  float* G = (float*)d_ws;                   // ~1.8 MB of operators + scan state

  build_block_ops<<<NBANDS, 32, 0, stream>>>(sos, G);
  iir_seg_pass<false><<<NBANDS * 2 * NSEG, 32, 0, stream>>>(x, G, (float*)d_out);
  combine_segments<<<NBANDS * 2, 32, 0, stream>>>(G);
  iir_seg_pass<true><<<NBANDS * 2 * NSEG, 32, 0, stream>>>(x, G, (float*)d_out);
}